// AdapativeStepLayer_52003464020333
// MI455X (gfx1250) — compile-verified
//
#include <hip/hip_runtime.h>
#include <hip/hip_bf16.h>
#include <math.h>

// ---------------------------------------------------------------------------
// Adaptive memory-network hop, MI455X (gfx1250, wave32).
//
// Roofline reasoning:
//  * Attention phase: per-batch GEMVs over encoded_knowledge (268 MB > 192 MB
//    L2) -> pure HBM bandwidth problem.  Online (flash) softmax reads
//    knowledge exactly once per hop: 8 * 268 MB = 2.1 GB -> ~92 us at
//    23.3 TB/s.  Tiles staged with GLOBAL_LOAD_ASYNC_TO_LDS_B128 (ASYNCcnt)
//    and double-buffered so HBM transfer of tile t+1 overlaps compute of
//    tile t.  Batch slices that fit the 192 MB L2 are loaded RT (resident
//    across hops); the overflow tail is loaded TH_LOAD_NT so it doesn't
//    thrash the resident set.
//  * Update GEMM [256 x 1024] @ [1024 x 512] has shared weights -> f32
//    matrix pipe: V_WMMA_F32_16X16X4_F32, W_update slab staged in LDS,
//    one wave per 16x16 C tile.
//  * Halting logic needs whole-batch ANY reductions -> tiny 1-block kernel;
//    hop-to-hop dependency is stream ordering (graph-capture safe).
// ---------------------------------------------------------------------------

#define BATCH 256
#define KDIM  512
#define DDIM  512
#define TILEK 16
#define ONE_MINUS_EPS 0.99f
#define MAX_COMP 8.0f

typedef __attribute__((ext_vector_type(2))) float v2f;
typedef __attribute__((ext_vector_type(8))) float v8f;

// ---------------------------------------------------------------------------
// init: mem <- current_memory ; mem_acc, att outputs <- 0 ; halting state init
// ---------------------------------------------------------------------------
__global__ __launch_bounds__(256) void init_kernel(
    const float* __restrict__ cur_mem,
    float* __restrict__ w_mem,
    float* __restrict__ out_memacc,
    float* __restrict__ out_att,
    float* __restrict__ w_acc, float* __restrict__ w_accc,
    float* __restrict__ w_hop, int* __restrict__ w_mask)
{
    int i = blockIdx.x * blockDim.x + threadIdx.x;     // 0 .. 131071
    w_mem[i]      = cur_mem[i];
    out_memacc[i] = 0.0f;
    out_att[i]    = 0.0f;
    if (i < BATCH) {
        w_acc[i]  = 0.0f;
        w_accc[i] = 0.0f;
        w_hop[i]  = 0.0f;
        w_mask[i] = 1;
    }
}

// ---------------------------------------------------------------------------
// Attention: one block (256 thr = 8 waves) per batch row.  Online softmax.
// k-tiles (16 x 512 f32 = 32 KB) staged via async global->LDS copies,
// double buffered (64 KB LDS) so the next tile streams from HBM while the
// current one is consumed.
// ---------------------------------------------------------------------------
__global__ __launch_bounds__(256) void attn_kernel(
    const float* __restrict__ q,
    const float* __restrict__ memv,
    const float* __restrict__ knowl,
    float* __restrict__ attended)
{
    __shared__ float v[DDIM];               // q + m
    __shared__ float kt[2][TILEK][DDIM];    // double-buffered tiles (64 KB)
    __shared__ float s_scores[TILEK];       // raw scores -> softmax weights
    __shared__ float s_M, s_S, s_scale;

    const int b    = blockIdx.x;
    const int tid  = threadIdx.x;
    const int lane = tid & 31;
    const int wave = tid >> 5;

    const float* kb = knowl + (size_t)b * KDIM * DDIM;
    // slices for b < 184 total ~192 MB -> keep RT (L2-resident across hops);
    // the overflow tail streams with a non-temporal hint.
    const bool resident = (b < 184);

    for (int d = tid; d < DDIM; d += 256)
        v[d] = q[b * DDIM + d] + memv[b * DDIM + d];
    if (tid == 0) { s_M = -3.0e38f; s_S = 0.0f; s_scale = 1.0f; }

    // Issue one tile's async copy: 256 lanes x 8 x b128 = 32 KB.
    auto issue_tile = [&](int k0, int bufsel) {
        unsigned long long g =
            (unsigned long long)(const void*)(kb + (size_t)k0 * DDIM) +
            (unsigned long long)tid * 16ull;
        unsigned l = (unsigned)(size_t)(&kt[bufsel][0][0]) + (unsigned)tid * 16u;
        if (resident) {
#pragma unroll
            for (int i = 0; i < 8; ++i)
                asm volatile("global_load_async_to_lds_b128 %0, %1, off"
                             :: "v"(l + 4096u * (unsigned)i),
                                "v"(g + 4096ull * (unsigned long long)i)
                             : "memory");
        } else {
#pragma unroll
            for (int i = 0; i < 8; ++i)
                asm volatile("global_load_async_to_lds_b128 %0, %1, off th:TH_LOAD_NT"
                             :: "v"(l + 4096u * (unsigned)i),
                                "v"(g + 4096ull * (unsigned long long)i)
                             : "memory");
        }
    };

    issue_tile(0, 0);

    float acc0 = 0.0f, acc1 = 0.0f;         // attended acc, d = tid, tid+256
    const int NT = KDIM / TILEK;            // 32 tiles

    for (int t = 0; t < NT; ++t) {
        const int cur = t & 1;

        // Prefetch next tile into the other buffer, then wait until this
        // wave's 8 issues for tile t have landed (loads complete in order).
        if (t + 1 < NT) {
            issue_tile((t + 1) * TILEK, cur ^ 1);
            asm volatile("s_wait_asynccnt 8" ::: "memory");
        } else {
            asm volatile("s_wait_asynccnt 0" ::: "memory");
        }
        __syncthreads();                    // all waves' tile-t data in LDS

        // ---- scores: wave w handles rows w and w+8 of the tile ----
#pragma unroll
        for (int r = 0; r < 2; ++r) {
            int tt = wave + 8 * r;
            float part = 0.0f;
#pragma unroll 4
            for (int d = lane; d < DDIM; d += 32)
                part += kt[cur][tt][d] * v[d];
            for (int off = 16; off > 0; off >>= 1)
                part += __shfl_down(part, off, 32);
            if (lane == 0) s_scores[tt] = part;
        }
        __syncthreads();

        // ---- online-softmax bookkeeping on wave 0 ----
        if (tid < 32) {
            float sc = (lane < TILEK) ? s_scores[lane] : -3.0e38f;
            float mx = sc;
            for (int off = 16; off > 0; off >>= 1)
                mx = fmaxf(mx, __shfl_down(mx, off, 32));
            mx = __shfl(mx, 0, 32);
            float newM = fmaxf(s_M, mx);
            float w = (lane < TILEK) ? __expf(sc - newM) : 0.0f;
            float sm = w;
            for (int off = 16; off > 0; off >>= 1)
                sm += __shfl_down(sm, off, 32);
            sm = __shfl(sm, 0, 32);
            if (lane < TILEK) s_scores[lane] = w;   // now holds exp-weights
            if (lane == 0) {
                float scale = __expf(s_M - newM);
                s_scale = scale;
                s_S = s_S * scale + sm;
                s_M = newM;
            }
        }
        __syncthreads();

        // ---- rescale + weighted accumulation from LDS tile ----
        float scale = s_scale;
        acc0 *= scale;
        acc1 *= scale;
#pragma unroll
        for (int tt = 0; tt < TILEK; ++tt) {
            float w = s_scores[tt];
            acc0 += w * kt[cur][tt][tid];
            acc1 += w * kt[cur][tt][tid + 256];
        }
        __syncthreads();   // also guards buffer cur from next iter's async writes
    }

    float invS = 1.0f / s_S;
    attended[b * DDIM + tid]       = acc0 * invS;
    attended[b * DDIM + tid + 256] = acc1 * invS;
}

// ---------------------------------------------------------------------------
// Update GEMM: new_m = tanh([m | attended] @ W_update), f32 WMMA 16x16x4.
// Grid: 32 blocks (one 16-col slab of W_update each).  Block: 512 thr =
// 16 waves; wave w owns output rows [16w, 16w+16).  W_update slab
// (1024 x 16 f32 = 64 KB) staged in LDS, reused by all 16 waves.
// VGPR layouts per CDNA5 ISA 7.12.2:
//   A 16x4 : lanes 0-15 hold (row=lane, K=0,1), lanes 16-31 (row=lane-16, K=2,3)
//   B 4x16 : VGPR0 = rows K=0 / K=2, VGPR1 = rows K=1 / K=3 (half-lane split)
//   C/D    : VGPR i -> M = i (lanes 0-15) / M = 8+i (lanes 16-31), N = lane&15
// ---------------------------------------------------------------------------
__global__ __launch_bounds__(512) void update_gemm_kernel(
    const float* __restrict__ memv,
    const float* __restrict__ attended,
    const float* __restrict__ W,        // [1024, 512] row-major
    float* __restrict__ new_m)          // [256, 512]
{
    __shared__ float lws[1024 * 16];    // W_update[:, n0:n0+16], 64 KB

    const int tid  = threadIdx.x;
    const int lane = tid & 31;
    const int wave = tid >> 5;          // 0..15
    const int n0   = blockIdx.x * 16;

    // stage the 16-wide weight slab
    for (int i = tid; i < 1024 * 16; i += 512) {
        int r = i >> 4, c = i & 15;
        lws[i] = W[r * 512 + n0 + c];
    }
    __syncthreads();

    const int b0   = wave * 16;
    const int rl   = lane & 15;
    const int half = (lane >> 4) << 1;  // 0 for lanes 0-15, 2 for lanes 16-31

    v8f c = {};

    // first 512 K-columns come from mem
    {
        const float* Arow = memv + (size_t)(b0 + rl) * DDIM;
#pragma unroll 4
        for (int kk = 0; kk < 512; kk += 4) {
            v2f a, bb;
            a[0]  = Arow[kk + half];
            a[1]  = Arow[kk + half + 1];
            bb[0] = lws[(kk + half) * 16 + rl];
            bb[1] = lws[(kk + half + 1) * 16 + rl];
            c = __builtin_amdgcn_wmma_f32_16x16x4_f32(
                    false, a, false, bb, (short)0, c, false, false);
        }
    }
    // second 512 K-columns come from attended
    {
        const float* Arow = attended + (size_t)(b0 + rl) * DDIM;
#pragma unroll 4
        for (int kk = 0; kk < 512; kk += 4) {
            v2f a, bb;
            a[0]  = Arow[kk + half];
            a[1]  = Arow[kk + half + 1];
            bb[0] = lws[(512 + kk + half) * 16 + rl];
            bb[1] = lws[(512 + kk + half + 1) * 16 + rl];
            c = __builtin_amdgcn_wmma_f32_16x16x4_f32(
                    false, a, false, bb, (short)0, c, false, false);
        }
    }

    // tanh + store per C/D layout
    const int mhi = (lane >> 4) << 3;   // +8 row shift for upper half-lanes
#pragma unroll
    for (int i = 0; i < 8; ++i) {
        int row = b0 + i + mhi;
        new_m[row * DDIM + n0 + rl] = tanhf(c[i]);
    }
}

// ---------------------------------------------------------------------------
// p[b] = sigmoid( new_m[b,:] . halting_weight )   (one wave per batch row)
// ---------------------------------------------------------------------------
__global__ __launch_bounds__(256) void halt_prob_kernel(
    const float* __restrict__ new_m,
    const float* __restrict__ hw,       // [512]
    float* __restrict__ p)
{
    const int lane = threadIdx.x & 31;
    const int wave = threadIdx.x >> 5;
    const int b = blockIdx.x * 8 + wave;
    const float* row = new_m + (size_t)b * DDIM;
    float s = 0.0f;
#pragma unroll 4
    for (int d = lane; d < DDIM; d += 32)
        s += row[d] * hw[d];
    for (int off = 16; off > 0; off >>= 1)
        s += __shfl_down(s, off, 32);
    if (lane == 0)
        p[b] = 1.0f / (1.0f + __expf(-s));
}

// ---------------------------------------------------------------------------
// Halting state update (single block, b = threadIdx.x); whole-batch ANYs.
// ---------------------------------------------------------------------------
__global__ __launch_bounds__(256) void halting_kernel(
    const float* __restrict__ p,
    float* __restrict__ acc, float* __restrict__ accc,
    float* __restrict__ hop, int* __restrict__ mask,
    float* __restrict__ upd, int* __restrict__ gate)
{
    __shared__ int s_any[2];
    const int b = threadIdx.x;

    float av = acc[b], cv = accc[b], hv = hop[b];
    int   mv = mask[b];
    float pv = p[b];

    if (b < 2) s_any[b] = 0;
    __syncthreads();
    if (cv < ONE_MINUS_EPS && hv < MAX_COMP) s_any[0] = 1;   // active (pre-hop)
    __syncthreads();
    int active = s_any[0];

    int   nmask = (av + pv < ONE_MINUS_EPS) && mv;
    float nf    = nmask ? 1.0f : 0.0f;
    float nacc  = av + pv * nf;
    float naccc = cv + pv * (mv ? 1.0f : 0.0f);
    float nhop  = hv + nf;

    if (nmask && nhop < MAX_COMP) s_any[1] = 1;              // cond (post-hop)
    __syncthreads();
    int cond = s_any[1];

    float u = cond ? pv * nf : (1.0f - pv);

    if (active) {
        acc[b]  = nacc;
        accc[b] = naccc;
        hop[b]  = nhop;
        mask[b] = nmask;
        upd[b]  = u;
    }
    if (b == 0) gate[0] = active;
}

// ---------------------------------------------------------------------------
// Gated state application: mem_acc += new_m*upd ; mem = new_m ; att = attended
// ---------------------------------------------------------------------------
__global__ __launch_bounds__(256) void apply_kernel(
    const float* __restrict__ new_m,
    const float* __restrict__ attended,
    const float* __restrict__ upd,
    const int* __restrict__ gate,
    float* __restrict__ memv,
    float* __restrict__ out_memacc,
    float* __restrict__ out_att)
{
    if (!gate[0]) return;               // whole hop was a no-op (scan gating)
    int i = blockIdx.x * blockDim.x + threadIdx.x;   // 0 .. 131071
    int b = i >> 9;                      // / DDIM
    float nm = new_m[i];
    out_memacc[i] += nm * upd[b];
    memv[i]        = nm;
    out_att[i]     = attended[i];
}

// ---------------------------------------------------------------------------
extern "C" void kernel_launch(void* const* d_in, const int* in_sizes, int n_in,
                              void* d_out, int out_size, void* d_ws, size_t ws_size,
                              hipStream_t stream)
{
    const float* q     = (const float*)d_in[0];   // [256, 512]
    const float* cmem  = (const float*)d_in[1];   // [256, 512]
    const float* knowl = (const float*)d_in[2];   // [256, 512, 512]
    const float* hw    = (const float*)d_in[3];   // [512, 1]
    const float* W     = (const float*)d_in[4];   // [1024, 512]

    const int NM = BATCH * DDIM;                  // 131072

    float* w_mem  = (float*)d_ws;
    float* w_att  = w_mem  + NM;
    float* w_newm = w_att  + NM;
    float* w_p    = w_newm + NM;
    float* w_acc  = w_p    + BATCH;
    float* w_accc = w_acc  + BATCH;
    float* w_hop  = w_accc + BATCH;
    float* w_upd  = w_hop  + BATCH;
    int*   w_mask = (int*)(w_upd + BATCH);
    int*   w_gate = w_mask + BATCH;

    float* out_memacc = (float*)d_out;            // [256, 512]
    float* out_att    = out_memacc + NM;          // [256, 512]

    init_kernel<<<NM / 256, 256, 0, stream>>>(
        cmem, w_mem, out_memacc, out_att, w_acc, w_accc, w_hop, w_mask);

    for (int hopi = 0; hopi < 8; ++hopi) {
        attn_kernel<<<BATCH, 256, 0, stream>>>(q, w_mem, knowl, w_att);
        update_gemm_kernel<<<DDIM / 16, 512, 0, stream>>>(w_mem, w_att, W, w_newm);
        halt_prob_kernel<<<BATCH / 8, 256, 0, stream>>>(w_newm, hw, w_p);
        halting_kernel<<<1, 256, 0, stream>>>(
            w_p, w_acc, w_accc, w_hop, w_mask, w_upd, w_gate);
        apply_kernel<<<NM / 256, 256, 0, stream>>>(
            w_newm, w_att, w_upd, w_gate, w_mem, out_memacc, out_att);
    }
}